// TFGuptaClassifier_47150150975961
// MI455X (gfx1250) — compile-verified
//
#include <hip/hip_runtime.h>
#include <math.h>

#define NQ   512        // queries
#define ND   27         // dims
#define DP   28         // padded dims (multiple of 4 for WMMA K)
#define NC   10         // classes
#define NN   200000     // features (exactly 12500 * 16 -> no OOB tiles)
#define NT   (NN / 16)  // 12500 feature tiles
#define WPQ  32         // waves per query tile
#define NBLK 128        // blocks for main kernel (128*8 = 1024 waves = 32 qt * 32 wq)
#define TPB  256
#define BIGF 1e30f

typedef __attribute__((ext_vector_type(2))) float v2f;
typedef __attribute__((ext_vector_type(8))) float v8f;

// ws layout (floats)
#define OF_SCALE 0
#define OF_QS    64
#define OF_QN    (OF_QS + NQ * DP)            // 14400
#define OF_FS    (OF_QN + NQ)                 // 14912
#define OF_FN    (OF_FS + NN * DP)            // 5614912
#define OF_CD    (OF_FN + NN)                 // 5814912
#define OF_CI    (OF_CD + NQ * WPQ * 3)       // 5864064
// total: 5913216 floats = ~23.7 MB

__device__ __forceinline__ void ins3(float v, int id,
                                     float &d0, float &d1, float &d2,
                                     int &i0, int &i1, int &i2) {
  if (v < d2) {
    d2 = v; i2 = id;
    if (d2 < d1) { float t = d1; d1 = d2; d2 = t; int ti = i1; i1 = i2; i2 = ti; }
    if (d1 < d0) { float t = d0; d0 = d1; d1 = t; int ti = i0; i0 = i1; i1 = ti; }
  }
}

// ---- k0: zero the scale accumulator -------------------------------------
__global__ void k0_init(float* scale) {
  if (threadIdx.x < 64) scale[threadIdx.x] = 0.0f;
}

// ---- k1: per-dim max-abs over features (bitwise atomicMax) --------------
__global__ void k1_scale(const float* __restrict__ feat, unsigned int* scale_bits) {
  __shared__ unsigned int smax[32];
  if (threadIdx.x < 32) smax[threadIdx.x] = 0u;
  __syncthreads();
  long long total = (long long)NN * ND;
  for (long long i = (long long)blockIdx.x * blockDim.x + threadIdx.x; i < total;
       i += (long long)gridDim.x * blockDim.x) {
    int d = (int)(i % ND);
    atomicMax(&smax[d], __float_as_uint(fabsf(feat[i])));  // |x|>=0: bit order == float order
  }
  __syncthreads();
  if (threadIdx.x < ND) atomicMax(&scale_bits[threadIdx.x], smax[threadIdx.x]);
}

// ---- k1b: scale + pad queries and features into ws (L2-resident) --------
__global__ void k1b_prescale(const float* __restrict__ qry,
                             const float* __restrict__ feat,
                             float* __restrict__ ws) {
  __shared__ float sc[32];
  if (threadIdx.x < 32) sc[threadIdx.x] = (threadIdx.x < ND) ? ws[OF_SCALE + threadIdx.x] : 1.0f;
  __syncthreads();
  float* qs = ws + OF_QS;
  float* fs = ws + OF_FS;
  long long total = (long long)(NQ + NN) * DP;
  for (long long i = (long long)blockIdx.x * blockDim.x + threadIdx.x; i < total;
       i += (long long)gridDim.x * blockDim.x) {
    if (i < (long long)NQ * DP) {
      int r = (int)(i / DP), d = (int)(i % DP);
      float o = 0.0f;
      if (d < ND) { float s = sc[d]; float v = qry[r * ND + d]; o = (s == 0.0f) ? 0.0f : v / s; }
      qs[i] = o;
    } else {
      long long j = i - (long long)NQ * DP;
      int r = (int)(j / DP), d = (int)(j % DP);
      float o = 0.0f;
      if (d < ND) { float s = sc[d]; float v = feat[(long long)r * ND + d]; o = (s == 0.0f) ? 0.0f : v / s; }
      fs[j] = o;
    }
  }
}

// ---- k1c: row norms of the scaled/padded arrays -------------------------
__global__ void k1c_norms(float* __restrict__ ws) {
  const float* qs = ws + OF_QS;
  const float* fs = ws + OF_FS;
  float* qn = ws + OF_QN;
  float* fn = ws + OF_FN;
  int total = NQ + NN;
  for (int r = blockIdx.x * blockDim.x + threadIdx.x; r < total;
       r += gridDim.x * blockDim.x) {
    const float* p = (r < NQ) ? (qs + (long long)r * DP) : (fs + (long long)(r - NQ) * DP);
    float s = 0.0f;
#pragma unroll
    for (int d = 0; d < DP; ++d) { float v = p[d]; s += v * v; }
    if (r < NQ) qn[r] = s; else fn[r - NQ] = s;
  }
}

// ---- k2: fp32 WMMA distances, per-lane register top-3 -------------------
__global__ __launch_bounds__(TPB) void k2_main(float* __restrict__ ws) {
  const float* qs = ws + OF_QS;
  const float* qn = ws + OF_QN;
  const float* fs = ws + OF_FS;
  const float* fn = ws + OF_FN;
  float* candd = ws + OF_CD;
  int*   candi = (int*)(ws + OF_CI);

  const int tid   = threadIdx.x;
  const int g     = blockIdx.x * (TPB / 32) + (tid >> 5);  // global wave id [0,1024)
  const int lane  = tid & 31;
  const int lhalf = lane >> 4;
  const int lcol  = lane & 15;
  const int qt    = g & 31;   // query tile this wave owns
  const int wq    = g >> 5;   // feature-stripe index [0,32)

  // A fragments for this wave's 16 queries (loaded once).
  // f32 A 16x4 layout: lane m = lane%16, K = vgpr + 2*(lane/16)
  const int qrow = qt * 16 + lcol;
  const float* qp = qs + qrow * DP;
  v2f af[7];
#pragma unroll
  for (int kk = 0; kk < 7; ++kk) {
    int k0 = 4 * kk + 2 * lhalf;
    v2f a; a.x = qp[k0]; a.y = qp[k0 + 1];
    af[kk] = a;
  }
  // query norms for the 8 C rows this lane sees: m = qt*16 + r + 8*lhalf
  float qnr[8];
#pragma unroll
  for (int r = 0; r < 8; ++r) qnr[r] = qn[qt * 16 + r + 8 * lhalf];

  // per-lane running top-3 (dist^2, feature idx) per C row
  float a0[8], a1[8], a2[8];
  int   b0[8], b1[8], b2[8];
#pragma unroll
  for (int r = 0; r < 8; ++r) {
    a0[r] = BIGF; a1[r] = BIGF; a2[r] = BIGF;
    b0[r] = -1;   b1[r] = -1;   b2[r] = -1;
  }

  // stream this wave's feature tiles (all loads hit the 192MB L2 after first pass)
  for (int t = wq; t < NT; t += WPQ) {
    const int col = t * 16 + lcol;           // feature row for this lane's B column
    const float* fp = fs + (long long)col * DP;
    // B 4x16 layout: lane n = lane%16, K = vgpr + 2*(lane/16) ; B[k][n] = fs[n][k]
    v2f bf[7];
#pragma unroll
    for (int kk = 0; kk < 7; ++kk) {
      int k0 = 4 * kk + 2 * lhalf;
      v2f b; b.x = fp[k0]; b.y = fp[k0 + 1];
      bf[kk] = b;
    }
    float fnorm = fn[col];

    v8f c = {0.f, 0.f, 0.f, 0.f, 0.f, 0.f, 0.f, 0.f};
#pragma unroll
    for (int kk = 0; kk < 7; ++kk)
      c = __builtin_amdgcn_wmma_f32_16x16x4_f32(false, af[kk], false, bf[kk],
                                                (short)0, c, false, false);
    // C/D: vgpr r, lane l -> M = r + 8*(l/16), N = l%16
#pragma unroll
    for (int r = 0; r < 8; ++r) {
      float d2 = qnr[r] + fnorm - 2.0f * c[r];
      // strict < keeps the earlier (lower) feature index on ties, like lax.top_k
      ins3(d2, col, a0[r], a1[r], a2[r], b0[r], b1[r], b2[r]);
    }
  }

  // one cross-lane extraction per wave: 3 smallest of the 16-lane union per row
#pragma unroll
  for (int r = 0; r < 8; ++r) {
    float v0 = a0[r], v1 = a1[r], v2 = a2[r];
    int   j0 = b0[r], j1 = b1[r], j2 = b2[r];
    float rd0, rd1, rd2; int ri0, ri1, ri2;
#pragma unroll
    for (int it = 0; it < 3; ++it) {
      float mv = v0; int mi = j0;
#pragma unroll
      for (int off = 1; off <= 8; off <<= 1) {
        float ov = __shfl_xor(mv, off, 32);
        int   oi = __shfl_xor(mi, off, 32);
        if (ov < mv || (ov == mv && oi < mi)) { mv = ov; mi = oi; }
      }
      // pop winner (feature indices are unique across lanes)
      if (j0 == mi) { v0 = v1; j0 = j1; v1 = v2; j1 = j2; v2 = BIGF; j2 = -1; }
      if (it == 0) { rd0 = mv; ri0 = mi; }
      else if (it == 1) { rd1 = mv; ri1 = mi; }
      else { rd2 = mv; ri2 = mi; }
    }
    if (lcol == 0) {   // leaders: lane 0 (rows m), lane 16 (rows m+8)
      int m = qt * 16 + r + 8 * lhalf;
      long long o = ((long long)m * WPQ + wq) * 3;
      candd[o + 0] = rd0; candd[o + 1] = rd1; candd[o + 2] = rd2;
      candi[o + 0] = ri0; candi[o + 1] = ri1; candi[o + 2] = ri2;
    }
  }
}

// ---- k3: final merge + weighted vote ------------------------------------
__global__ void k3_final(const float* __restrict__ labels,
                         const float* __restrict__ ws,
                         float* __restrict__ out) {
  const float* candd = ws + OF_CD;
  const int*   candi = (const int*)(ws + OF_CI);
  int q = blockIdx.x * blockDim.x + threadIdx.x;
  if (q >= NQ) return;
  float d0 = BIGF, d1 = BIGF, d2 = BIGF;
  int i0 = 0, i1 = 0, i2 = 0;
  for (int wq = 0; wq < WPQ; ++wq) {
    long long o = ((long long)q * WPQ + wq) * 3;
    ins3(candd[o + 0], candi[o + 0], d0, d1, d2, i0, i1, i2);
    ins3(candd[o + 1], candi[o + 1], d0, d1, d2, i0, i1, i2);
    ins3(candd[o + 2], candi[o + 2], d0, d1, d2, i0, i1, i2);
  }
  float kd0 = sqrtf(fmaxf(d0, 0.0f));
  float kd1 = sqrtf(fmaxf(d1, 0.0f));
  float kd2 = sqrtf(fmaxf(d2, 0.0f));
  float s0 = (kd0 == 0.0f) ? 1.0f : kd0;
  float s1 = (kd1 == 0.0f) ? 1.0f : kd1;
  float s2 = (kd2 == 0.0f) ? 1.0f : kd2;
  float lab0[NC], wsum[NC];
  for (int cc = 0; cc < NC; ++cc) {
    float l0 = labels[(long long)i0 * NC + cc];
    float l1 = labels[(long long)i1 * NC + cc];
    float l2 = labels[(long long)i2 * NC + cc];
    lab0[cc] = l0;
    wsum[cc] = l0 / s0 + l1 / s1 + l2 / s2;
  }
  int am = 0;
  for (int cc = 1; cc < NC; ++cc)
    if (wsum[cc] > wsum[am]) am = cc;   // first max, like jnp.argmax
  bool zero_hit = (kd0 == 0.0f);        // kd sorted ascending
  float* o = out + (long long)q * (3 + NC);
  o[0] = kd0; o[1] = kd1; o[2] = kd2;
  for (int cc = 0; cc < NC; ++cc)
    o[3 + cc] = zero_hit ? lab0[cc] : ((cc == am) ? 1.0f : 0.0f);
}

extern "C" void kernel_launch(void* const* d_in, const int* in_sizes, int n_in,
                              void* d_out, int out_size, void* d_ws, size_t ws_size,
                              hipStream_t stream) {
  const float* qry    = (const float*)d_in[0];  // (512, 27)
  const float* feat   = (const float*)d_in[1];  // (200000, 27)
  const float* labels = (const float*)d_in[2];  // (200000, 10)
  float* ws = (float*)d_ws;

  k0_init     <<<1,    64,  0, stream>>>(ws + OF_SCALE);
  k1_scale    <<<256,  TPB, 0, stream>>>(feat, (unsigned int*)(ws + OF_SCALE));
  k1b_prescale<<<1024, TPB, 0, stream>>>(qry, feat, ws);
  k1c_norms   <<<784,  TPB, 0, stream>>>(ws);
  k2_main     <<<NBLK, TPB, 0, stream>>>(ws);
  k3_final    <<<2,    TPB, 0, stream>>>(labels, ws, (float*)d_out);
}